// GeneralizedAdvantageEstimation_62208306315435
// MI455X (gfx1250) — compile-verified
//
#include <hip/hip_runtime.h>

// GAE (reverse linear-recurrence scan), bandwidth-optimal 3-pass chunked scan.
// g[t] = delta[t] + c[t]*g[t+1],  c[t] = gamma*lambda*(1-done[t])
// Per chunk the map g_in = B + A*g_out composes linearly -> parallel scan.
//
// All indexing is 32-bit (tensor is 33.5 MB < 2^31 bytes) so the hot loops
// carry a single int element-offset decremented by N, giving saddr+voffset
// addressing for every stream instead of per-iteration 64-bit MADs.

typedef __attribute__((ext_vector_type(4))) float v4f;
typedef __attribute__((ext_vector_type(4))) int   v4i;

namespace {
constexpr float kGamma  = 0.99f;
constexpr float kLambda = 0.95f;
constexpr int   kBlock  = 256;   // 8 waves of 32 on CDNA5
constexpr int   kEPT    = 4;     // envs per thread (float4 / b128 path)
}

// WRITE=0 (pass 1): chunk-local reverse scan with g=0; emit A=prod(c), B=partial g.
// WRITE=1 (pass 3): chunk-local reverse scan seeded with Gin; emit adv and returns.
template <int WRITE>
__global__ __launch_bounds__(kBlock)
void gae_chunk_kernel(const float* __restrict__ rew,
                      const float* __restrict__ val,
                      const float* __restrict__ nval,
                      const int*   __restrict__ done,
                      float* __restrict__ Aout,
                      float* __restrict__ Bout,
                      const float* __restrict__ Gin,
                      float* __restrict__ adv,
                      float* __restrict__ ret,
                      int N, int L)
{
    const int chunk = blockIdx.y;
    const int e = (blockIdx.x * kBlock + threadIdx.x) * kEPT;
    if (e >= N) return;

    const float gd  = kGamma * kLambda;
    const v4f   one = (v4f){1.f, 1.f, 1.f, 1.f};

    v4f g, a;
    if (WRITE) {
        g = *reinterpret_cast<const v4f*>(Gin + chunk * N + e);
        a = one;                         // unused in this instantiation
    } else {
        g = (v4f){0.f, 0.f, 0.f, 0.f};
        a = one;
    }

    // 32-bit element offset of row tHi for this thread's 4 envs.
    int off = (chunk * L + (L - 1)) * N + e;

#pragma unroll 2
    for (int i = 0; i < L; ++i, off -= N) {
        // Prefetch 8 rows ahead in scan order (8 KB stride; -64KB folds into
        // the 24-bit signed instruction offset; speculative OOB is dropped).
        __builtin_prefetch(rew  + (off - 8 * N), 0, 3);
        __builtin_prefetch(val  + (off - 8 * N), 0, 3);
        __builtin_prefetch(nval + (off - 8 * N), 0, 3);
        __builtin_prefetch(done + (off - 8 * N), 0, 3);

        v4f r, v, nv; v4i d;
        if (WRITE) {
            // Last use of the inputs: non-temporal so the streaming output
            // stores don't fight the (L2-resident) input set.
            r  = __builtin_nontemporal_load(reinterpret_cast<const v4f*>(rew  + off));
            v  = __builtin_nontemporal_load(reinterpret_cast<const v4f*>(val  + off));
            nv = __builtin_nontemporal_load(reinterpret_cast<const v4f*>(nval + off));
            d  = __builtin_nontemporal_load(reinterpret_cast<const v4i*>(done + off));
        } else {
            // First pass: default RT caching to warm the 192MB L2 for pass 3.
            r  = *reinterpret_cast<const v4f*>(rew  + off);
            v  = *reinterpret_cast<const v4f*>(val  + off);
            nv = *reinterpret_cast<const v4f*>(nval + off);
            d  = *reinterpret_cast<const v4i*>(done + off);
        }

        const v4f nd    = one - __builtin_convertvector(d, v4f);
        const v4f c     = gd * nd;
        const v4f delta = r + kGamma * nv * nd - v;
        g = delta + c * g;
        if (!WRITE) a *= c;

        if (WRITE) {
            __builtin_nontemporal_store(g,     reinterpret_cast<v4f*>(adv + off));
            __builtin_nontemporal_store(g + v, reinterpret_cast<v4f*>(ret + off));
        }
    }

    if (!WRITE) {
        const int coff = chunk * N + e;
        *reinterpret_cast<v4f*>(Aout + coff) = a;
        *reinterpret_cast<v4f*>(Bout + coff) = g;
    }
}

// Pass 2: per env, reverse-combine the C chunk summaries to get each chunk's
// incoming g. C*N FMAs total -> negligible.
__global__ __launch_bounds__(kBlock)
void gae_combine_kernel(const float* __restrict__ A,
                        const float* __restrict__ B,
                        float* __restrict__ Gin,
                        int N, int C)
{
    const int e = (blockIdx.x * kBlock + threadIdx.x) * kEPT;
    if (e >= N) return;
    v4f carry = (v4f){0.f, 0.f, 0.f, 0.f};
    int off = (C - 1) * N + e;           // 32-bit offsets here too
    for (int k = 0; k < C; ++k, off -= N) {
        *reinterpret_cast<v4f*>(Gin + off) = carry;   // g entering this chunk
        const v4f a = *reinterpret_cast<const v4f*>(A + off);
        const v4f b = *reinterpret_cast<const v4f*>(B + off);
        carry = b + a * carry;
    }
}

extern "C" void kernel_launch(void* const* d_in, const int* in_sizes, int n_in,
                              void* d_out, int out_size, void* d_ws, size_t ws_size,
                              hipStream_t stream)
{
    const float* rew  = (const float*)d_in[0];
    const float* val  = (const float*)d_in[1];
    const float* nval = (const float*)d_in[2];
    const int*   done = (const int*)d_in[3];

    const int N = 2048;                  // N_ENVS (reference constant)
    const int T = in_sizes[0] / N;       // 4096

    // Chunk count: 64 gives 32K threads in the heavy passes; shrink if the
    // workspace is tight or T isn't divisible.
    int C = 64;
    while (C > 1 && ((T % C) != 0 ||
                     3ull * (size_t)C * (size_t)N * sizeof(float) > ws_size))
        C >>= 1;
    const int L = T / C;

    float* A   = (float*)d_ws;
    float* B   = A + (size_t)C * N;
    float* Gin = B + (size_t)C * N;

    float* adv = (float*)d_out;          // output tuple: (adv, returns) concat
    float* ret = adv + (size_t)T * N;

    const int envThreads = N / kEPT;                           // 512
    const int envBlocks  = (envThreads + kBlock - 1) / kBlock; // 2
    dim3 block(kBlock);
    dim3 grid1(envBlocks, C);

    gae_chunk_kernel<0><<<grid1, block, 0, stream>>>(
        rew, val, nval, done, A, B, nullptr, nullptr, nullptr, N, L);

    gae_combine_kernel<<<envBlocks, block, 0, stream>>>(A, B, Gin, N, C);

    gae_chunk_kernel<1><<<grid1, block, 0, stream>>>(
        rew, val, nval, done, nullptr, nullptr, Gin, adv, ret, N, L);
}